// OFTV2_45028437131814
// MI455X (gfx1250) — compile-verified
//
#include <hip/hip_runtime.h>
#include <cstdint>

#define NCAM 6
#define CCH  64
#define FH   32
#define FW   88
#define HC   7
#define DG   128

// LDS layout (single shared block => base offset 0):
//   [0   .. 191]  camera params (6 x 32 floats), filled by async global->LDS
//   [192 .. 959]  BEV grid tile (2 rows x 384 floats), filled by TDM
#define SP_WORDS   (NCAM * 32)
#define SG_WORDS   (2 * 384)
#define SG_BYTEOFF (SP_WORDS * 4)

// 2-wide float vector with 4-byte alignment: lets the backend merge the two
// adjacent bilinear taps of each feature row into one (unaligned) b64 load.
typedef float v2f __attribute__((ext_vector_type(2), aligned(4)));

// Vector types for the TDM descriptor builtin (probe-verified signatures).
typedef unsigned tdm_u32x4 __attribute__((ext_vector_type(4)));
typedef int      tdm_i32x4 __attribute__((ext_vector_type(4)));
typedef int      tdm_i32x8 __attribute__((ext_vector_type(8)));

// ---------------------------------------------------------------------------
// MAIN (first in file: histogram/snippet cover a prefix of the disasm).
// One thread per (b, hc, d, w) BEV cell; loops cameras, gathers 64 channels,
// max-reduces in registers, streams out with non-temporal stores.
// block = (128, 2): w = tid.x, d = 2*bid.x + tid.y; grid = (64, 7, 2)
// ---------------------------------------------------------------------------
__global__ __launch_bounds__(256)
void oft_main_kernel(const float* __restrict__ feat,
                     const float* __restrict__ bevgrid,
                     const float* __restrict__ pbuf,
                     float* __restrict__ out) {
  __shared__ float smem[SP_WORDS + SG_WORDS];

  const int w  = threadIdx.x;                   // 0..127
  const int d0 = blockIdx.x * 2;                // first of 2 d-rows
  const int d  = d0 + threadIdx.y;              // 0..127
  const int hc = blockIdx.y;                    // 0..6
  const int b  = blockIdx.z;                    // 0..1
  const int t  = threadIdx.y * 128 + threadIdx.x;

  // --- TDM: one wave DMAs the 2x384-float grid tile (row stride 387) ------
  if (t == 0) {   // EXEC ignored by tensor ops; only wave0 reaches this block
    const uint64_t ga = (uint64_t)(uintptr_t)(
        bevgrid + ((size_t)b * 129 + d0) * 129 * 3);
    tdm_u32x4 g0;
    g0[0] = 1u;                                  // count=1 valid descriptor
    g0[1] = (unsigned)SG_BYTEOFF;                // lds_addr (bytes)
    g0[2] = (unsigned)(ga & 0xffffffffu);        // global_addr[31:0]
    g0[3] = (unsigned)((ga >> 32) & 0x01ffffffu) // global_addr[56:32]
            | 0x80000000u;                       // type=2 ("image") bits127:126
    tdm_i32x8 g1;
    g1[0] = (int)(2u << 16);                     // data_size=2 -> 4 bytes
    g1[1] = (int)(387u << 16);                   // tensor_dim0 lo16 = 387
    g1[2] = (int)(2u << 16);                     // dim0 hi=0 | tensor_dim1=2
    g1[3] = (int)(384u << 16);                   // dim1 hi=0 | tile_dim0=384
    g1[4] = 2;                                   // tile_dim1=2, tile_dim2=0
    g1[5] = 387;                                 // tensor_dim0_stride lo32
    g1[6] = 0;                                   // stride hi / dim1_stride lo
    g1[7] = 0;
    tdm_i32x4 gz = {0, 0, 0, 0};
#if defined(__clang_major__) && __clang_major__ >= 23
    tdm_i32x8 gz8 = {0, 0, 0, 0, 0, 0, 0, 0};
    __builtin_amdgcn_tensor_load_to_lds(g0, g1, gz, gz, gz8, 0);
#else
    __builtin_amdgcn_tensor_load_to_lds(g0, g1, gz, gz, 0);
#endif
    __builtin_amdgcn_s_wait_tensorcnt(0);
  }

  // --- Async global->LDS broadcast of the camera parameter block ----------
  if (t < SP_WORDS) {
    unsigned lds_off = (unsigned)(t * 4);
    uint64_t gaddr =
        (uint64_t)(uintptr_t)(pbuf + (size_t)b * SP_WORDS + t);
    asm volatile("global_load_async_to_lds_b32 %0, %1, off"
                 :: "v"(lds_off), "v"(gaddr) : "memory");
  }
  asm volatile("s_wait_asynccnt 0x0" ::: "memory");
  __syncthreads();

  // BEV corner from the TDM-staged LDS tile
  const float* sg = smem + SP_WORDS + threadIdx.y * 384 + w * 3;
  const float c0 = sg[0];
  const float c1 = sg[1] + (2.0f - 0.5f * (float)hc);
  const float c2 = sg[2];

  float acc[CCH];
#pragma unroll
  for (int c = 0; c < CCH; ++c) acc[c] = -__builtin_inff();

  for (int n = 0; n < NCAM; ++n) {
    const float* p = &smem[n * 32];
    // perspective projection
    float hx = p[0] * c0 + p[1] * c1 + p[2]  * c2 + p[3];
    float hy = p[4] * c0 + p[5] * c1 + p[6]  * c2 + p[7];
    float hz = p[8] * c0 + p[9] * c1 + p[10] * c2 + p[11];
    const float m = (hz > 0.f) ? 1.f : 0.f;
    hx *= m; hy *= m;
    const float px = hx / hz, py = hy / hz;
    const float fx = p[12], fy = p[13], cxx = p[14], cyy = p[15];
    const float x  = (px - cxx) / fx;
    const float yv = (py - cyy) / fy;
    const float r2 = x * x + yv * yv;
    float dxv, dyv;
    if (p[22] == 1.0f) {  // fisheye (uniform per camera within wave)
      const float r  = sqrtf(r2);
      const float th = atanf(r);
      const float t2 = th * th, t4 = t2 * t2;
      const float rad =
          th * (1.f + p[16] * t2 + p[17] * t4 + p[18] * t4 * t2 +
                p[21] * t4 * t4) / r;
      dxv = x  * rad * fx + cxx;
      dyv = yv * rad * fy + cyy;
    } else {              // pinhole radial+tangential
      const float radial = 1.f + p[16] * r2 + p[17] * r2 * r2 +
                           p[18] * r2 * r2 * r2;
      const float xp = x  * radial + 2.f * p[19] * x * yv +
                       p[20] * (r2 + 2.f * x * x);
      const float yp = yv * radial + p[19] * (r2 + 2.f * yv * yv) +
                       2.f * p[20] * x * yv;
      dxv = xp * fx + cxx;
      dyv = yp * fy + cyy;
    }
    dxv *= m; dyv *= m;
    // post-augmentation affine, normalize, clip
    const float tx = p[24] * dxv + p[25] * dyv + p[28];
    const float ty = p[26] * dxv + p[27] * dyv + p[29];
    const float nx = fminf(fmaxf(tx * (1.f / 44.f) - 1.f, -1.f), 1.f);
    const float ny = fminf(fmaxf(ty * (1.f / 44.f) - 1.f, -1.f), 1.f);
    const bool vis = (nx > -1.f) && (nx < 1.f) && (ny > -1.f) && (ny < 1.f);
    if (!vis) {
#pragma unroll
      for (int c = 0; c < CCH; ++c) acc[c] = fmaxf(acc[c], 0.f);
      continue;
    }
    // bilinear sample (align_corners=True); visible => no clamping needed:
    // gx in (0,87), gy in (0,31) so x0<=86, y0<=30 always.
    const float gx = (nx + 1.f) * 43.5f;   // (W-1)/2
    const float gy = (ny + 1.f) * 15.5f;   // (H-1)/2
    const float x0f = floorf(gx), y0f = floorf(gy);
    const float wx = gx - x0f, wy = gy - y0f;
    const int   xi = (int)x0f, yi = (int)y0f;
    const float w00 = (1.f - wx) * (1.f - wy), w01 = wx * (1.f - wy);
    const float w10 = (1.f - wx) * wy,         w11 = wx * wy;
    const float* fb =
        feat + (((size_t)(b * NCAM + n) * CCH) * FH + yi) * FW + xi;
    __builtin_prefetch(fb, 0, 1);   // global_prefetch_b8
#pragma unroll
    for (int c = 0; c < CCH; ++c) {
      const float* f0 = fb + c * (FH * FW);
      const v2f r0 = *(const v2f*)(f0);        // taps (y0,x0),(y0,x1)
      const v2f r1 = *(const v2f*)(f0 + FW);   // taps (y1,x0),(y1,x1)
      const float v = r0.x * w00 + r0.y * w01 + r1.x * w10 + r1.y * w11;
      acc[c] = fmaxf(acc[c], v);
    }
  }

  // out[b][c*7+hc][d][w], streamed (non-temporal) to keep features in L2
  float* ob = out + (((size_t)b * (CCH * HC) + hc) * DG + d) * DG + w;
#pragma unroll
  for (int c = 0; c < CCH; ++c)
    __builtin_nontemporal_store(acc[c], ob + (size_t)c * HC * DG * DG);
}

// ---------------------------------------------------------------------------
// Setup: pack per-(b,n) camera parameters into a 32-float record in d_ws.
//   [0..11]  calib = ks @ imu2cs (3x4 row-major)
//   [12..15] fx, fy, cx, cy
//   [16..22] undists d0..d6 (d6 = fisheye flag)
//   [24..27] post_rot r00,r01,r10,r11   [28..29] post_trans tx,ty
// ---------------------------------------------------------------------------
__global__ void oft_setup_kernel(const float* __restrict__ ks,
                                 const float* __restrict__ imu,
                                 const float* __restrict__ prot,
                                 const float* __restrict__ ptrn,
                                 const float* __restrict__ und,
                                 float* __restrict__ pbuf) {
  int i = threadIdx.x;              // (b*NCAM + n)
  if (i >= 2 * NCAM) return;
  const float* K = ks  + i * 9;
  const float* M = imu + i * 12;
  float*       P = pbuf + i * 32;
#pragma unroll
  for (int r = 0; r < 3; ++r)
#pragma unroll
    for (int c = 0; c < 4; ++c) {
      float s = 0.f;
#pragma unroll
      for (int k = 0; k < 3; ++k) s += K[r * 3 + k] * M[k * 4 + c];
      P[r * 4 + c] = s;
    }
  P[12] = K[0]; P[13] = K[4]; P[14] = K[2]; P[15] = K[5];
#pragma unroll
  for (int j = 0; j < 7; ++j) P[16 + j] = und[i * 7 + j];
  P[23] = 0.f;
  const float* R = prot + i * 9;
  P[24] = R[0]; P[25] = R[1]; P[26] = R[3]; P[27] = R[4];
  P[28] = ptrn[i * 3 + 0]; P[29] = ptrn[i * 3 + 1];
  P[30] = 0.f; P[31] = 0.f;
}

// ---------------------------------------------------------------------------
extern "C" void kernel_launch(void* const* d_in, const int* in_sizes, int n_in,
                              void* d_out, int out_size, void* d_ws,
                              size_t ws_size, hipStream_t stream) {
  const float* features   = (const float*)d_in[0];
  const float* ks         = (const float*)d_in[1];
  const float* imu2cs     = (const float*)d_in[2];
  const float* post_rots  = (const float*)d_in[3];
  const float* post_trans = (const float*)d_in[4];
  const float* undists    = (const float*)d_in[5];
  const float* bevgrid    = (const float*)d_in[6];
  // d_in[7] = drop_idx (==NCAM -> no camera dropped; reference ignores it)

  float* pbuf = (float*)d_ws;  // 2*6*32 floats = 1.5 KB

  oft_setup_kernel<<<1, 16, 0, stream>>>(ks, imu2cs, post_rots, post_trans,
                                         undists, pbuf);

  dim3 blk(128, 2, 1);
  dim3 grd(DG / 2, HC, 2);
  oft_main_kernel<<<grd, blk, 0, stream>>>(features, bevgrid, pbuf,
                                           (float*)d_out);
}